// StandardSelfAttention_79929341378755
// MI455X (gfx1250) — compile-verified
//
#include <hip/hip_runtime.h>
#include <math.h>
#include <stdint.h>

typedef float v2f __attribute__((ext_vector_type(2)));
typedef float v8f __attribute__((ext_vector_type(8)));
typedef unsigned int u32x4 __attribute__((ext_vector_type(4)));
typedef int i32x4 __attribute__((ext_vector_type(4)));
typedef int i32x8 __attribute__((ext_vector_type(8)));

#define BB 2
#define SS 2048
#define DM 1024
#define NH 16
#define HD 64

__device__ __forceinline__ v8f wmma_f32(v2f a, v2f b, v8f c) {
  // D = A(16x4) * B(4x16) + C(16x16), fp32
  return __builtin_amdgcn_wmma_f32_16x16x4_f32(
      /*neg_a=*/false, a, /*neg_b=*/false, b,
      /*c_mod=*/(short)0, c, /*reuse_a=*/false, /*reuse_b=*/false);
}

// ---------------------------------------------------------------------------
// TDM: DMA a 2D tile (rows_tile x 16 fp32, row stride = stride0 elements)
// from global memory into LDS at lds_byte_off. LDS padding: +2 DWORDs after
// every 16 DWORDs -> 18-float LDS row stride (bank-conflict-free frag reads).
// D# per CDNA5 ISA; tracked on TENSORcnt.
// This toolchain uses the 6-arg builtin: (g0, g1, g2, g3, g4, cpol).
// ---------------------------------------------------------------------------
__device__ __forceinline__ void tdm_load_2d(const float* gptr,
                                            unsigned lds_byte_off,
                                            int rows_tile, unsigned tensor_d0,
                                            unsigned tensor_d1,
                                            unsigned stride0) {
  const unsigned long long ga = (unsigned long long)(uintptr_t)gptr;
  const unsigned galo = (unsigned)__builtin_amdgcn_readfirstlane((int)(ga & 0xFFFFFFFFu));
  const unsigned gahi = (unsigned)__builtin_amdgcn_readfirstlane((int)(ga >> 32));
  const unsigned ldo  = (unsigned)__builtin_amdgcn_readfirstlane((int)lds_byte_off);

  u32x4 g0;
  g0[0] = 1u;                                   // count=1 (valid descriptor)
  g0[1] = ldo;                                  // lds_addr (bytes)
  g0[2] = galo;                                 // global_addr[31:0]
  g0[3] = (gahi & 0x01FFFFFFu) | (2u << 30);    // global_addr[56:32] | type=2

  i32x8 g1;
  // data_size=4B (2<<16) | pad_enable (1<<20) | pad_interval=16DW (3<<22)
  // | pad_amount=2DW (1<<25)
  g1[0] = (int)((2u << 16) | (1u << 20) | (3u << 22) | (1u << 25));
  g1[1] = (int)((tensor_d0 & 0xFFFFu) << 16);                       // dim0 lo
  g1[2] = (int)(((tensor_d0 >> 16) & 0xFFFFu) |
                ((tensor_d1 & 0xFFFFu) << 16));                     // dim0 hi | dim1 lo
  g1[3] = (int)(((tensor_d1 >> 16) & 0xFFFFu) | (16u << 16));       // dim1 hi | tile_dim0=16
  g1[4] = rows_tile & 0xFFFF;                                       // tile_dim1
  g1[5] = (int)stride0;                                             // dim0_stride lo32
  g1[6] = 0;                                                        // dim0_stride hi | dim1_stride lo
  g1[7] = 0;
  const i32x4 z4 = (i32x4){0, 0, 0, 0};
  const i32x8 z8 = (i32x8){0, 0, 0, 0, 0, 0, 0, 0};
  __builtin_amdgcn_tensor_load_to_lds(g0, g1, z4, z4, z8, 0);
}

// ---------------------------------------------------------------------------
// C[M,N] = A[M,K] @ W[N,K]^T + bias[N]
// Block = 4 waves; each wave computes 32(M) x 64(N): 2 M-tiles x 4 N-tiles.
// TDM double-buffers a 32x16 A slice + 64x16 W slice per wave into LDS
// (18-float padded rows); s_wait_tensorcnt(2) retires the previous stage
// while the next stage's DMA is in flight. wmma reads fragments via ds_load.
// Fragment layouts:
//   A-frag: lane l, vgpr v -> A[m*16 + l%16][k0 + 2*(l/16) + v]
//   B-frag: lane l, vgpr v -> W[nt*16 + l%16][k0 + 2*(l/16) + v]
//   D:      lane l, vgpr r -> C[mt*16 + r + 8*(l/16)][nt*16 + l%16]
// ---------------------------------------------------------------------------
#define ISSUE_STAGE(k0, offA, offW)                                           \
  do {                                                                        \
    tdm_load_2d(Abase + (size_t)(k0), (unsigned)(uintptr_t)&lds[(offA)], 32,  \
                (unsigned)K, (unsigned)M, (unsigned)K);                       \
    tdm_load_2d(Wbase + (size_t)(k0), (unsigned)(uintptr_t)&lds[(offW)], 64,  \
                (unsigned)K, (unsigned)N, (unsigned)K);                       \
  } while (0)

#define COMPUTE_STAGE(offA, offW)                                             \
  do {                                                                        \
    _Pragma("unroll") for (int i = 0; i < 4; ++i) {                           \
      const int ko = i * 4 + 2 * hi;                                          \
      v2f a0 = *(const v2f*)&lds[(offA) + ml * 18 + ko];                      \
      v2f a1 = *(const v2f*)&lds[(offA) + (16 + ml) * 18 + ko];               \
      _Pragma("unroll") for (int j = 0; j < 4; ++j) {                         \
        v2f bw = *(const v2f*)&lds[(offW) + (j * 16 + ml) * 18 + ko];         \
        acc[0][j] = wmma_f32(a0, bw, acc[0][j]);                              \
        acc[1][j] = wmma_f32(a1, bw, acc[1][j]);                              \
      }                                                                       \
    }                                                                         \
  } while (0)

__global__ __launch_bounds__(128) void gemm_bias_wmma(
    const float* __restrict__ A, const float* __restrict__ W,
    const float* __restrict__ bias, float* __restrict__ C,
    int M, int N, int K) {
  const int lane = threadIdx.x & 31;
  const int warp = threadIdx.x >> 5;
  const int ml = lane & 15;
  const int hi = lane >> 4;

  const int mt0 = (blockIdx.y * 4 + warp) * 2;  // first of 2 M-tiles
  const int nt0 = blockIdx.x * 4;               // first of 4 N-tiles

  // per-wave LDS staging: A 32x18 x2 stages, W 64x18 x2 stages (floats)
  __shared__ float lds[4 * 3456];
  const int wbase = warp * 3456;
  const int offA0 = wbase, offA1 = wbase + 576;
  const int offW0 = wbase + 1152, offW1 = wbase + 2304;

  const float* Abase = A + (size_t)(mt0 * 16) * K;
  const float* Wbase = W + (size_t)(nt0 * 16) * K;

  v8f acc[2][4];
#pragma unroll
  for (int m = 0; m < 2; ++m)
#pragma unroll
    for (int j = 0; j < 4; ++j) acc[m][j] = (v8f){};

  ISSUE_STAGE(0, offA0, offW0);
  ISSUE_STAGE(16, offA1, offW1);

  for (int kc = 0; kc < K; kc += 32) {
    __builtin_amdgcn_s_wait_tensorcnt(2);  // stage0 (kc) landed in LDS
    COMPUTE_STAGE(offA0, offW0);
    asm volatile("s_wait_dscnt 0x0" ::: "memory");  // drain reads before DMA overwrite
    const int kn0 = (kc + 32 < K) ? (kc + 32) : kc;
    ISSUE_STAGE(kn0, offA0, offW0);

    __builtin_amdgcn_s_wait_tensorcnt(2);  // stage1 (kc+16) landed in LDS
    COMPUTE_STAGE(offA1, offW1);
    asm volatile("s_wait_dscnt 0x0" ::: "memory");
    const int kn1 = (kc + 48 < K) ? (kc + 48) : (kc + 16);
    ISSUE_STAGE(kn1, offA1, offW1);
  }

#pragma unroll
  for (int m = 0; m < 2; ++m) {
#pragma unroll
    for (int j = 0; j < 4; ++j) {
      const int n = (nt0 + j) * 16 + ml;
      const float bv = bias[n];
      float* Cp = C + (size_t)((mt0 + m) * 16) * N + n;
#pragma unroll
      for (int r = 0; r < 8; ++r)
        Cp[(size_t)(r + 8 * hi) * N] = acc[m][j][r] + bv;
    }
  }
}

// ---------------------------------------------------------------------------
// Split qkv (B,S,3*DM) into Q/K/V (B,H,S,HD), applying RoPE to Q and K.
// ---------------------------------------------------------------------------
__global__ __launch_bounds__(256) void rope_split(
    const float* __restrict__ qkv, float* __restrict__ Q,
    float* __restrict__ K, float* __restrict__ V) {
  const int t = blockIdx.x * blockDim.x + threadIdx.x;  // 21 bits total
  const int j = t & 31;
  const int s = (t >> 5) & (SS - 1);
  const int h = (t >> 16) & (NH - 1);
  const int b = t >> 20;

  const size_t src = ((size_t)(b * SS + s)) * (3 * DM) + h * HD;
  const float q1 = qkv[src + j],          q2 = qkv[src + j + 32];
  const float k1 = qkv[src + DM + j],     k2 = qkv[src + DM + j + 32];
  const float v1 = qkv[src + 2 * DM + j], v2 = qkv[src + 2 * DM + j + 32];

  const float inv = __powf(10000.0f, -(float)j * (1.0f / 32.0f));
  const float ang = (float)s * inv;
  float sn, cs;
  __sincosf(ang, &sn, &cs);

  const size_t dst = ((size_t)((b * NH + h) * SS + s)) * HD;
  Q[dst + j]      = q1 * cs - q2 * sn;
  Q[dst + j + 32] = q1 * sn + q2 * cs;
  K[dst + j]      = k1 * cs - k2 * sn;
  K[dst + j + 32] = k1 * sn + k2 * cs;
  V[dst + j]      = v1;
  V[dst + j + 32] = v2;
}

// ---------------------------------------------------------------------------
// Flash attention, causal. One wave per (b, h, 16-query tile).
// Pipelined: K-fragments of tile kt+1 load during softmax/P.V of tile kt;
// V-fragments batch-issued ahead of the P.V wmma chain; next V tile prefetched.
// ---------------------------------------------------------------------------
__global__ __launch_bounds__(32) void flash_attn(
    const float* __restrict__ Q, const float* __restrict__ K,
    const float* __restrict__ V, float* __restrict__ Y) {
  const int lane = threadIdx.x;
  const int ml = lane & 15;
  const int hi = lane >> 4;
  const int mt = blockIdx.x;
  const int h  = blockIdx.y;
  const int b  = blockIdx.z;

  const size_t head = (size_t)(b * NH + h) * SS * HD;
  const float* Qh = Q + head;
  const float* Kh = K + head;
  const float* Vh = V + head;

  // Preload Q A-fragments: qa[i] covers K-dim [4i, 4i+4)
  v2f qa[16];
  const int qrow = mt * 16 + ml;
#pragma unroll
  for (int i = 0; i < 16; ++i)
    qa[i] = *(const v2f*)&Qh[(size_t)qrow * HD + i * 4 + 2 * hi];

  v8f oacc[4];
#pragma unroll
  for (int c = 0; c < 4; ++c) oacc[c] = (v8f){};
  float mstat[8], lstat[8];
#pragma unroll
  for (int r = 0; r < 8; ++r) { mstat[r] = -1e30f; lstat[r] = 0.0f; }

  __shared__ float pl[16 * 17];

  // K-fragment double buffer, preload tile 0
  v2f kbC[16], kbN[16];
#pragma unroll
  for (int i = 0; i < 16; ++i)
    kbC[i] = *(const v2f*)&Kh[(size_t)ml * HD + i * 4 + 2 * hi];

  for (int kt = 0; kt <= mt; ++kt) {
    const float* Vb = Vh + (size_t)kt * 16 * HD;

    // S tile = Q @ K^T (consumes current K fragments)
    v8f sc = (v8f){};
#pragma unroll
    for (int i = 0; i < 16; ++i) sc = wmma_f32(qa[i], kbC[i], sc);

    const bool have_next = (kt < mt);  // wave-uniform
    if (have_next) {
      const float* Kn = Kh + (size_t)(kt + 1) * 16 * HD;
#pragma unroll
      for (int i = 0; i < 16; ++i)
        kbN[i] = *(const v2f*)&Kn[(size_t)ml * HD + i * 4 + 2 * hi];
      // prefetch next V tile (global_prefetch_b8); 32 lanes x 128B = 4KB tile
      __builtin_prefetch(Vh + (size_t)(kt + 1) * 16 * HD + lane * 32, 0, 1);
    }

    // batch-issue V fragment loads for this tile (hide under softmax math)
    float vf[4][4][2];
#pragma unroll
    for (int c = 0; c < 4; ++c)
#pragma unroll
      for (int i = 0; i < 4; ++i) {
        const int k0 = i * 4 + 2 * hi;
        vf[c][i][0] = Vb[(size_t)(k0 + 0) * HD + c * 16 + ml];
        vf[c][i][1] = Vb[(size_t)(k0 + 1) * HD + c * 16 + ml];
      }

    // scale + causal mask (row = r + 8*hi, col = ml in D-layout)
    float sv[8];
    const int kj = kt * 16 + ml;
#pragma unroll
    for (int r = 0; r < 8; ++r) {
      const int qi = mt * 16 + r + 8 * hi;
      float v = sc[r] * 0.125f;  // 1/sqrt(64)
      sv[r] = (kj > qi) ? -1e30f : v;
    }

    // row max (a row's 16 values live in one half-wave -> xor masks 1,2,4,8)
    float mnew[8], alpha[8];
#pragma unroll
    for (int r = 0; r < 8; ++r) {
      float v = sv[r];
      v = fmaxf(v, __shfl_xor(v, 1, 32));
      v = fmaxf(v, __shfl_xor(v, 2, 32));
      v = fmaxf(v, __shfl_xor(v, 4, 32));
      v = fmaxf(v, __shfl_xor(v, 8, 32));
      mnew[r]  = fmaxf(mstat[r], v);
      alpha[r] = __expf(mstat[r] - mnew[r]);
      mstat[r] = mnew[r];
      sv[r]    = __expf(sv[r] - mnew[r]);
    }
    // row sum + rescale running state
#pragma unroll
    for (int r = 0; r < 8; ++r) {
      float v = sv[r];
      v += __shfl_xor(v, 1, 32);
      v += __shfl_xor(v, 2, 32);
      v += __shfl_xor(v, 4, 32);
      v += __shfl_xor(v, 8, 32);
      lstat[r] = lstat[r] * alpha[r] + v;
    }
#pragma unroll
    for (int c = 0; c < 4; ++c)
#pragma unroll
      for (int r = 0; r < 8; ++r) oacc[c][r] *= alpha[r];

    // restripe P (D-layout) -> A-fragments via LDS
    __syncthreads();
#pragma unroll
    for (int r = 0; r < 8; ++r) pl[(r + 8 * hi) * 17 + ml] = sv[r];
    __syncthreads();

    v2f pa[4];
#pragma unroll
    for (int i = 0; i < 4; ++i) {
      const int k0 = i * 4 + 2 * hi;
      pa[i] = (v2f){pl[ml * 17 + k0], pl[ml * 17 + k0 + 1]};
    }

    // O += P @ V  (V fragments already resident)
#pragma unroll
    for (int c = 0; c < 4; ++c)
#pragma unroll
      for (int i = 0; i < 4; ++i) {
        v2f vb = (v2f){vf[c][i][0], vf[c][i][1]};
        oacc[c] = wmma_f32(pa[i], vb, oacc[c]);
      }

    // rotate K double buffer
    if (have_next) {
#pragma unroll
      for (int i = 0; i < 16; ++i) kbC[i] = kbN[i];
    }
  }

  // normalize + store to Y (B,S,DM) with heads concatenated
  float* Yp = Y + ((size_t)b * SS + mt * 16) * DM + h * HD;
#pragma unroll
  for (int r = 0; r < 8; ++r) {
    const float inv = 1.0f / lstat[r];
#pragma unroll
    for (int c = 0; c < 4; ++c)
      Yp[(size_t)(r + 8 * hi) * DM + c * 16 + ml] = oacc[c][r] * inv;
  }
}

// ---------------------------------------------------------------------------
extern "C" void kernel_launch(void* const* d_in, const int* in_sizes, int n_in,
                              void* d_out, int out_size, void* d_ws, size_t ws_size,
                              hipStream_t stream) {
  (void)in_sizes; (void)n_in; (void)out_size; (void)ws_size;
  const float* x     = (const float*)d_in[0];
  const float* qkv_w = (const float*)d_in[1];
  const float* qkv_b = (const float*)d_in[2];
  const float* out_w = (const float*)d_in[3];
  const float* out_b = (const float*)d_in[4];
  float* out = (float*)d_out;

  float* ws  = (float*)d_ws;
  float* qkv = ws;                                       // B*S*3*DM
  float* Qb  = qkv + (size_t)BB * SS * 3 * DM;           // B*H*S*HD
  float* Kb  = Qb + (size_t)BB * NH * SS * HD;
  float* Vb  = Kb + (size_t)BB * NH * SS * HD;
  float* Yb  = Vb + (size_t)BB * NH * SS * HD;           // B*S*DM

  const int Mrows = BB * SS;  // 4096

  // 1) qkv = x @ qkv_w^T + qkv_b      (M=4096, N=3072, K=1024)
  gemm_bias_wmma<<<dim3((3 * DM) / 64, Mrows / 128), 128, 0, stream>>>(
      x, qkv_w, qkv_b, qkv, Mrows, 3 * DM, DM);

  // 2) RoPE + split into Q,K,V (B,H,S,HD)
  rope_split<<<(BB * NH * SS * 32) / 256, 256, 0, stream>>>(qkv, Qb, Kb, Vb);

  // 3) causal flash attention -> Y (B,S,DM)
  flash_attn<<<dim3(SS / 16, NH, BB), 32, 0, stream>>>(Qb, Kb, Vb, Yb);

  // 4) out = Y @ out_w^T + out_b      (M=4096, N=1024, K=1024)
  gemm_bias_wmma<<<dim3(DM / 64, Mrows / 128), 128, 0, stream>>>(
      Yb, out_w, out_b, out, Mrows, DM, DM);
}